// GNNREG_11476152615620
// MI455X (gfx1250) — compile-verified
//
#include <hip/hip_runtime.h>
#include <math.h>

// ---------------------------------------------------------------------------
// GNN TransformerConv pipeline for gfx1250 (MI455X), wave32.
// GEMMs: V_WMMA_F32_16X16X4_F32, weights staged in LDS (<=32KB chunks),
// one wave per 16-row stripe computing the full output width (A-frag reuse).
// Edge softmax/aggregation via wave-per-edge kernels + L2 atomics.
// ---------------------------------------------------------------------------

typedef __attribute__((ext_vector_type(2))) float v2f;
typedef __attribute__((ext_vector_type(8))) float v8f;

#define WAVE 32

__device__ __forceinline__ float wave_sum(float v) {
#pragma unroll
    for (int o = 16; o; o >>= 1) v += __shfl_xor(v, o, WAVE);
    return v;
}

// Monotonic-int-encoding float atomic max (works regardless of sign).
__device__ __forceinline__ void atomicMaxF(float* addr, float val) {
    if (val >= 0.0f) {
        atomicMax((int*)addr, __float_as_int(val));
    } else {
        atomicMin((unsigned int*)addr, __float_as_uint(val));
    }
}

// ---------------------------------------------------------------------------
// fill kernel (scratch init; d_ws is poisoned by the harness)
// ---------------------------------------------------------------------------
__global__ void __launch_bounds__(256) k_fill(float* __restrict__ p, float v, size_t n) {
    size_t i = (size_t)blockIdx.x * blockDim.x + threadIdx.x;
    if (i < n) p[i] = v;
}

// ---------------------------------------------------------------------------
// Y[N,M] = X[N,K] @ W[K,M] + bias[M]     (N%16==0; K,M compile-time)
//
// Block = 256 threads = 8 waves. Each wave owns one 16-row stripe of X and
// computes ALL M columns (TN = M/16 accumulator tiles), so the A fragment is
// loaded once per K-step and reused TN times. W is cooperatively staged into
// LDS in KC-row chunks (<=32KB) so B fragments come from ds_load and W hits
// HBM once per workgroup.
//
// WMMA fragment layouts (ISA 7.12.2, f32 16x16x4):
//   A: lanes 0-15 hold row=lane {K,K+1}; lanes 16-31 hold row=lane-16 {K+2,K+3}
//   B: vgpr0 = rows {K | K+2} across lane halves, vgpr1 = rows {K+1 | K+3}
//   C/D: vgpr j -> row (hi*8 + j), col = lane&15
// ---------------------------------------------------------------------------
template <int K, int M>
__global__ void __launch_bounds__(256)
k_gemm_stripe(const float* __restrict__ X, const float* __restrict__ W,
              const float* __restrict__ bias, float* __restrict__ Y, int N) {
    constexpr int TN = M / 16;                       // column tiles per stripe
    constexpr int KC = (8192 / M) < K ? (8192 / M) : K;  // rows of W per LDS chunk
    __shared__ float sW[KC * M];                     // <= 32 KB

    const int lane   = threadIdx.x & (WAVE - 1);
    const int l15    = lane & 15;
    const int hi     = lane >> 4;                    // 0: K,K+1   1: K+2,K+3
    const int stripe = blockIdx.x * 8 + (threadIdx.x >> 5);
    const bool active = (stripe < (N >> 4));         // wave-uniform predicate
    const int row    = stripe * 16 + l15;
    const float* __restrict__ xr = X + (size_t)row * K;

    v8f acc[TN];
#pragma unroll
    for (int t = 0; t < TN; ++t) acc[t] = (v8f){0.f,0.f,0.f,0.f,0.f,0.f,0.f,0.f};

    for (int kc = 0; kc < K; kc += KC) {
        __syncthreads();
        // stage W[kc .. kc+KC) into LDS, float4-vectorized, all 256 threads
        {
            const float4* __restrict__ wsrc = (const float4*)(W + (size_t)kc * M);
            float4* __restrict__ wdst = (float4*)sW;
            constexpr int TOT4 = (KC * M) / 4;
#pragma unroll
            for (int i = 0; i < TOT4 / 256; ++i)
                wdst[i * 256 + threadIdx.x] = wsrc[i * 256 + threadIdx.x];
        }
        __syncthreads();

        if (active) {                                // EXEC all-1s inside (wave-uniform)
            for (int k0 = 0; k0 < KC; k0 += 4) {
                const int ka = k0 + hi * 2;
                v2f a;
                a.x = xr[kc + ka];                   // contiguous pair -> b64 load
                a.y = xr[kc + ka + 1];
#pragma unroll
                for (int t = 0; t < TN; ++t) {
                    v2f b;
                    b.x = sW[ka * M + t * 16 + l15];
                    b.y = sW[(ka + 1) * M + t * 16 + l15];
                    acc[t] = __builtin_amdgcn_wmma_f32_16x16x4_f32(
                                 /*neg_a=*/false, a, /*neg_b=*/false, b,
                                 /*c_mod=*/(short)0, acc[t],
                                 /*reuse_a=*/false, /*reuse_b=*/false);
                }
            }
        }
    }

    if (active) {
        const int rbase = stripe * 16 + hi * 8;
#pragma unroll
        for (int t = 0; t < TN; ++t) {
            const int col = t * 16 + l15;
            const float bb = bias[col];
            float* yb = Y + (size_t)rbase * M + col;
#pragma unroll
            for (int j = 0; j < 8; ++j)
                yb[(size_t)j * M] = acc[t][j] + bb;
        }
    }
}

// ---------------------------------------------------------------------------
// Wave per edge: score[e,h] = dot(q[dst], k[src]) per head / 8, seg-max into smax
// Row layout: head0 = cols [0,64), head1 = cols [64,128).
// ---------------------------------------------------------------------------
__global__ void __launch_bounds__(256)
k_edge_score(const float* __restrict__ q, const float* __restrict__ kk,
             const int* __restrict__ src, const int* __restrict__ dst,
             float* __restrict__ score, float* __restrict__ smax, int E) {
    const int lane = threadIdx.x & (WAVE - 1);
    const int e    = blockIdx.x * (blockDim.x >> 5) + (threadIdx.x >> 5);
    if (e >= E) return;
    const int s = src[e], d = dst[e];
    const float* __restrict__ qr = q + (size_t)d * 128;
    const float* __restrict__ kr = kk + (size_t)s * 128;
    float s0 = qr[lane] * kr[lane] + qr[lane + 32] * kr[lane + 32];
    float s1 = qr[64 + lane] * kr[64 + lane] + qr[96 + lane] * kr[96 + lane];
    s0 = wave_sum(s0);
    s1 = wave_sum(s1);
    if (lane == 0) {
        s0 *= 0.125f;                       // 1/sqrt(64)
        s1 *= 0.125f;
        score[2 * (size_t)e]     = s0;
        score[2 * (size_t)e + 1] = s1;
        atomicMaxF(&smax[2 * (size_t)d],     s0);
        atomicMaxF(&smax[2 * (size_t)d + 1], s1);
    }
}

// ---------------------------------------------------------------------------
// Wave per edge: w = exp(score - smax[dst]); denom[dst] += w;
// agg[dst] += w * v[src]   (unnormalized; normalized in node finalize)
// ---------------------------------------------------------------------------
__global__ void __launch_bounds__(256)
k_edge_accum(const float* __restrict__ v, const int* __restrict__ src,
             const int* __restrict__ dst, const float* __restrict__ score,
             const float* __restrict__ smax, float* __restrict__ denom,
             float* __restrict__ agg, int E) {
    const int lane = threadIdx.x & (WAVE - 1);
    const int e    = blockIdx.x * (blockDim.x >> 5) + (threadIdx.x >> 5);
    if (e >= E) return;
    const int s = src[e], d = dst[e];
    const float w0 = __expf(score[2 * (size_t)e]     - smax[2 * (size_t)d]);
    const float w1 = __expf(score[2 * (size_t)e + 1] - smax[2 * (size_t)d + 1]);
    if (lane == 0) {
        atomicAdd(&denom[2 * (size_t)d],     w0);
        atomicAdd(&denom[2 * (size_t)d + 1], w1);
    }
    const float* __restrict__ vr = v + (size_t)s * 128;
    float* __restrict__ ar = agg + (size_t)d * 128;
    atomicAdd(&ar[lane],       w0 * vr[lane]);
    atomicAdd(&ar[lane + 32],  w0 * vr[lane + 32]);
    atomicAdd(&ar[64 + lane],  w1 * vr[64 + lane]);
    atomicAdd(&ar[96 + lane],  w1 * vr[96 + lane]);
}

// ---------------------------------------------------------------------------
// Wave per node: out = relu(mean_heads(agg/denom) + skip); optional LayerNorm(64)
// ---------------------------------------------------------------------------
__global__ void __launch_bounds__(256)
k_node_finalize(const float* __restrict__ agg, const float* __restrict__ denom,
                const float* __restrict__ skip, const float* __restrict__ g,
                const float* __restrict__ beta, float* __restrict__ h,
                int N, int doLN) {
    const int lane = threadIdx.x & (WAVE - 1);
    const int n    = blockIdx.x * (blockDim.x >> 5) + (threadIdx.x >> 5);
    if (n >= N) return;
    const float d0 = denom[2 * (size_t)n];
    const float d1 = denom[2 * (size_t)n + 1];
    const float r0 = (d0 > 0.f) ? (1.0f / d0) : 0.f;   // empty segment -> 0
    const float r1 = (d1 > 0.f) ? (1.0f / d1) : 0.f;
    const float* __restrict__ ar = agg + (size_t)n * 128;
    const float* __restrict__ sr = skip + (size_t)n * 64;
    float oa = 0.5f * (ar[lane]      * r0 + ar[64 + lane] * r1) + sr[lane];
    float ob = 0.5f * (ar[lane + 32] * r0 + ar[96 + lane] * r1) + sr[lane + 32];
    oa = fmaxf(oa, 0.f);
    ob = fmaxf(ob, 0.f);
    if (doLN) {
        const float mu = wave_sum(oa + ob) * (1.0f / 64.0f);
        const float da = oa - mu, db = ob - mu;
        const float var = wave_sum(da * da + db * db) * (1.0f / 64.0f);
        const float rs  = rsqrtf(var + 1e-5f);
        oa = da * rs * g[lane]      + beta[lane];
        ob = db * rs * g[lane + 32] + beta[lane + 32];
    }
    h[(size_t)n * 64 + lane]      = oa;
    h[(size_t)n * 64 + lane + 32] = ob;
}

// ---------------------------------------------------------------------------
// Wave per node: pooled[batch[n]] += h[n]; cnt[batch[n]] += 1
// ---------------------------------------------------------------------------
__global__ void __launch_bounds__(256)
k_pool_accum(const float* __restrict__ h, const int* __restrict__ batch,
             float* __restrict__ pooled, float* __restrict__ cnt, int N) {
    const int lane = threadIdx.x & (WAVE - 1);
    const int n    = blockIdx.x * (blockDim.x >> 5) + (threadIdx.x >> 5);
    if (n >= N) return;
    const int gph = batch[n];
    atomicAdd(&pooled[(size_t)gph * 64 + lane],      h[(size_t)n * 64 + lane]);
    atomicAdd(&pooled[(size_t)gph * 64 + lane + 32], h[(size_t)n * 64 + lane + 32]);
    if (lane == 0) atomicAdd(&cnt[gph], 1.0f);
}

// ---------------------------------------------------------------------------
// Wave per graph: out[g] = dot(pooled[g]/max(cnt,1), Wp) + bp
// ---------------------------------------------------------------------------
__global__ void __launch_bounds__(256)
k_final_proj(const float* __restrict__ pooled, const float* __restrict__ cnt,
             const float* __restrict__ Wp, const float* __restrict__ bp,
             float* __restrict__ out, int G) {
    const int lane = threadIdx.x & (WAVE - 1);
    const int gph  = blockIdx.x * (blockDim.x >> 5) + (threadIdx.x >> 5);
    if (gph >= G) return;
    const float c = fmaxf(cnt[gph], 1.0f);
    float s = pooled[(size_t)gph * 64 + lane]      * Wp[lane]
            + pooled[(size_t)gph * 64 + lane + 32] * Wp[lane + 32];
    s = wave_sum(s);
    if (lane == 0) out[gph] = s / c + bp[0];
}

// ---------------------------------------------------------------------------
// host-side launch
// ---------------------------------------------------------------------------
static inline int cdiv(long long a, long long b) { return (int)((a + b - 1) / b); }

extern "C" void kernel_launch(void* const* d_in, const int* in_sizes, int n_in,
                              void* d_out, int out_size, void* d_ws, size_t ws_size,
                              hipStream_t stream) {
    const int IN_DIM = 128, HID = 64, G = 64;
    const int N = in_sizes[0] / IN_DIM;        // 50000 (multiple of 16)
    const int E = in_sizes[1] / 2;             // 800000

    const float* x     = (const float*)d_in[0];
    const int*   src   = (const int*)d_in[1];
    const int*   dst   = src + E;
    const int*   batch = (const int*)d_in[2];
    const float* g0    = (const float*)d_in[27];
    const float* be0   = (const float*)d_in[28];
    const float* g1    = (const float*)d_in[29];
    const float* be1   = (const float*)d_in[30];
    const float* Wp    = (const float*)d_in[31];
    const float* bp    = (const float*)d_in[32];

    // scratch layout (floats); agg aliases qbuf (q is dead after score kernel)
    float*       ws    = (float*)d_ws;
    const size_t N128  = (size_t)N * 128;
    const size_t N64   = (size_t)N * 64;
    float* qbuf  = ws;                  // [N,128]  (reused as agg)
    float* kbuf  = qbuf + N128;         // [N,128]
    float* vbuf  = kbuf + N128;         // [N,128]
    float* skip  = vbuf + N128;         // [N,64]
    float* score = skip + N64;          // [E,2]
    float* smax  = score + (size_t)E*2; // [N,2]
    float* denom = smax + (size_t)N*2;  // [N,2]
    float* hbuf  = denom + (size_t)N*2; // [N,64]
    float* pooled= hbuf + N64;          // [G,64]
    float* cnt   = pooled + (size_t)G*64; // [G]

    const int BLK = 256, WPB = BLK / 32;
    const int edgeBlocks = cdiv(E, WPB);
    const int nodeBlocks = cdiv(N, WPB);
    const int gemmBlocks = cdiv(N / 16, 8);    // one wave per 16-row stripe

    auto gemm = [&](const float* X, const float* W, const float* b, float* Y,
                    int din, int M) {
        if (din == 128 && M == 128)
            k_gemm_stripe<128, 128><<<gemmBlocks, BLK, 0, stream>>>(X, W, b, Y, N);
        else if (din == 128 && M == 64)
            k_gemm_stripe<128, 64><<<gemmBlocks, BLK, 0, stream>>>(X, W, b, Y, N);
        else if (din == 64 && M == 128)
            k_gemm_stripe<64, 128><<<gemmBlocks, BLK, 0, stream>>>(X, W, b, Y, N);
        else
            k_gemm_stripe<64, 64><<<gemmBlocks, BLK, 0, stream>>>(X, W, b, Y, N);
    };

    const float* hin = x;
    int din = IN_DIM;
    for (int l = 0; l < 3; ++l) {
        const float* Wq = (const float*)d_in[3 + l * 8 + 0];
        const float* bq = (const float*)d_in[3 + l * 8 + 1];
        const float* Wk = (const float*)d_in[3 + l * 8 + 2];
        const float* bk = (const float*)d_in[3 + l * 8 + 3];
        const float* Wv = (const float*)d_in[3 + l * 8 + 4];
        const float* bv = (const float*)d_in[3 + l * 8 + 5];
        const float* Ws = (const float*)d_in[3 + l * 8 + 6];
        const float* bs = (const float*)d_in[3 + l * 8 + 7];

        // node GEMMs (WMMA + LDS-staged weights)
        gemm(hin, Wq, bq, qbuf, din, 128);
        gemm(hin, Wk, bk, kbuf, din, 128);
        gemm(hin, Wv, bv, vbuf, din, 128);
        gemm(hin, Ws, bs, skip, din, 64);

        // segment softmax state
        k_fill<<<cdiv((size_t)N * 2, BLK), BLK, 0, stream>>>(smax, -3.402823466e38f, (size_t)N * 2);
        k_fill<<<cdiv((size_t)N * 2, BLK), BLK, 0, stream>>>(denom, 0.0f, (size_t)N * 2);

        // edge scores + segment max
        k_edge_score<<<edgeBlocks, BLK, 0, stream>>>(qbuf, kbuf, src, dst, score, smax, E);

        // agg (aliases qbuf; q is dead now) = 0, then unnormalized accumulate
        k_fill<<<cdiv(N128, BLK), BLK, 0, stream>>>(qbuf, 0.0f, N128);
        k_edge_accum<<<edgeBlocks, BLK, 0, stream>>>(vbuf, src, dst, score, smax, denom, qbuf, E);

        // head mean + skip + relu (+ LN for layers 0,1)
        const int doLN = (l < 2) ? 1 : 0;
        const float* gg = (l == 0) ? g0 : g1;
        const float* bb = (l == 0) ? be0 : be1;
        k_node_finalize<<<nodeBlocks, BLK, 0, stream>>>(qbuf, denom, skip, gg, bb, hbuf, N, doLN);

        hin = hbuf;
        din = HID;
    }

    // global mean pool + projection
    k_fill<<<cdiv((size_t)G * 64 + G, BLK), BLK, 0, stream>>>(pooled, 0.0f, (size_t)G * 64 + G);
    k_pool_accum<<<nodeBlocks, BLK, 0, stream>>>(hbuf, batch, pooled, cnt, N);
    k_final_proj<<<cdiv(G, WPB), BLK, 0, stream>>>(pooled, cnt, Wp, bp, (float*)d_out, G);
}